// VectorizedHypertoroidalSplat_43224550867596
// MI455X (gfx1250) — compile-verified
//
#include <hip/hip_runtime.h>
#include <math.h>

// Shapes from the reference: b=8, s=2048, d=512, E=8
#define BATCH 8
#define SEQ   2048
#define DIM   512
#define NE    8
#define KP    12          // rank 9 padded to 12 = 3 x K=4 WMMA steps
#define EPSV  1e-8f

typedef __attribute__((ext_vector_type(2))) float v2f;
typedef __attribute__((ext_vector_type(8))) float v8f;

// ---------------------------------------------------------------------------
// Stage 1: per-token features.
//   slot 0        : surface weight (gauss or torus branch, runtime-selected)
//   slots 1..8    : exp(-0.5*||x - p_e||^2)  (query side additionally * coef_e)
//   slots 9..11   : 0 (K padding)
// One wave (32 lanes) per token; lane-strided reduction over d=512.
// ---------------------------------------------------------------------------
__global__ __launch_bounds__(256)
void hts_feat_kernel(const float* __restrict__ Q,
                     const float* __restrict__ Kx,
                     const float* __restrict__ center,
                     const float* __restrict__ outer_radius,
                     const float* __restrict__ hole_radius,
                     const float* __restrict__ entry_points,
                     const float* __restrict__ entry_strengths,
                     const float* __restrict__ entry_orient,
                     float* __restrict__ FQ,   // [BATCH*SEQ, KP]
                     float* __restrict__ FK)   // [BATCH*SEQ, KP]
{
    const int wave = threadIdx.x >> 5;
    const int lane = threadIdx.x & 31;
    const int NT   = BATCH * SEQ;

    int t = blockIdx.x * 8 + wave;            // token id in [0, 2*NT)
    const bool isQ = (t < NT);
    const int  tok = isQ ? t : (t - NT);
    const float* x = (isQ ? Q : Kx) + (size_t)tok * DIM;

    float acc_c = 0.0f;        // sum (x - center)^2
    float accp[NE];
    #pragma unroll
    for (int e = 0; e < NE; ++e) accp[e] = 0.0f;

    for (int i = lane; i < DIM; i += 32) {
        float xv = x[i];
        float dc = xv - center[i];
        acc_c = fmaf(dc, dc, acc_c);
        #pragma unroll
        for (int e = 0; e < NE; ++e) {
            float dp = xv - entry_points[e * DIM + i];
            accp[e] = fmaf(dp, dp, accp[e]);
        }
    }

    // wave32 butterfly reduction
    #pragma unroll
    for (int off = 16; off > 0; off >>= 1) {
        acc_c += __shfl_xor(acc_c, off, 32);
        #pragma unroll
        for (int e = 0; e < NE; ++e)
            accp[e] += __shfl_xor(accp[e], off, 32);
    }

    if (lane == 0) {
        float R  = outer_radius[0];
        float hr = hole_radius[0] / (R + EPSV);
        float dist = sqrtf(acc_c);
        float w;
        if (hr < 0.05f) {                 // gaussian branch
            float z = dist / (R + EPSV);
            w = __expf(-0.5f * z * z);
        } else {                          // torus-shell branch
            float z = fabsf(dist - R);
            w = __expf(-0.5f * z * z);
        }
        float* dst = (isQ ? FQ : FK) + (size_t)tok * KP;
        dst[0] = w;
        #pragma unroll
        for (int e = 0; e < NE; ++e) {
            float we = __expf(-0.5f * accp[e]);
            if (isQ) {
                float sg = 1.0f / (1.0f + __expf(-entry_strengths[e]));
                float cf = (sg > 0.3f) ? sg * tanhf(entry_orient[e]) : 0.0f;
                we *= cf;
            }
            dst[1 + e] = we;
        }
        dst[9]  = 0.0f;
        dst[10] = 0.0f;
        dst[11] = 0.0f;
    }
}

// ---------------------------------------------------------------------------
// Stage 2: out[b] = FQ[b] (2048 x 12) * FK[b]^T (12 x 2048) via
// V_WMMA_F32_16X16X4_F32, 3 K-steps per 16x16 tile.
// One wave per 16(M) x 128(N) strip: A-frags loaded once, 8 B tiles.
//
// A (16x4 f32) layout: lanes 0-15 -> M=lane, VGPR0/1 = K=0/1;
//                      lanes 16-31 -> M=lane-16, VGPR0/1 = K=2/3.
// B (4x16) mirrors with N in place of M.
// C/D: VGPR v -> M = v (+8 for lanes 16-31), N = lane&15.
// EXEC is all ones (no divergence in this kernel).
// ---------------------------------------------------------------------------
__global__ __launch_bounds__(256)
void hts_gemm_kernel(const float* __restrict__ FQ,
                     const float* __restrict__ FK,
                     float* __restrict__ out)
{
    const int wave = threadIdx.x >> 5;
    const int lane = threadIdx.x & 31;

    // strips: BATCH * (SEQ/16) * (SEQ/128) = 8*128*16 = 16384; 8 waves/block
    int sid = blockIdx.x * 8 + wave;
    int b   = sid >> 11;            // / (128*16)
    int rem = sid & 2047;
    int m0  = (rem >> 4) << 4;      // row tile base, 0..2032
    int n0  = (rem & 15) << 7;      // col strip base, 0..1920

    const float* Aq = FQ + (size_t)b * SEQ * KP;
    const float* Bk = FK + (size_t)b * SEQ * KP;
    float*       O  = out + (size_t)b * SEQ * SEQ;

    const int l15   = lane & 15;
    const int khalf = (lane >> 4) * 2;     // 0 or 2

    const int arow = m0 + l15;
    v2f a0 = *(const v2f*)(Aq + (size_t)arow * KP + 0 + khalf);
    v2f a1 = *(const v2f*)(Aq + (size_t)arow * KP + 4 + khalf);
    v2f a2 = *(const v2f*)(Aq + (size_t)arow * KP + 8 + khalf);

    const int rbase = m0 + ((lane >> 4) << 3);

    #pragma unroll 2
    for (int j = 0; j < 8; ++j) {
        int ncol = n0 + j * 16 + l15;
        v2f b0 = *(const v2f*)(Bk + (size_t)ncol * KP + 0 + khalf);
        v2f b1 = *(const v2f*)(Bk + (size_t)ncol * KP + 4 + khalf);
        v2f b2 = *(const v2f*)(Bk + (size_t)ncol * KP + 8 + khalf);

        v8f c = {};
        c = __builtin_amdgcn_wmma_f32_16x16x4_f32(false, a0, false, b0,
                                                  (short)0, c, false, false);
        c = __builtin_amdgcn_wmma_f32_16x16x4_f32(false, a1, false, b1,
                                                  (short)0, c, false, false);
        c = __builtin_amdgcn_wmma_f32_16x16x4_f32(false, a2, false, b2,
                                                  (short)0, c, false, false);

        #pragma unroll
        for (int v = 0; v < 8; ++v) {
            O[(size_t)(rbase + v) * SEQ + ncol] = c[v];
        }
    }
}

// ---------------------------------------------------------------------------
extern "C" void kernel_launch(void* const* d_in, const int* in_sizes, int n_in,
                              void* d_out, int out_size, void* d_ws, size_t ws_size,
                              hipStream_t stream) {
    const float* Q    = (const float*)d_in[0];   // [8,2048,512]
    const float* Kx   = (const float*)d_in[1];   // [8,2048,512]
    const float* ctr  = (const float*)d_in[2];   // [512]
    const float* orad = (const float*)d_in[3];   // [1]
    const float* hrad = (const float*)d_in[4];   // [1]
    const float* epts = (const float*)d_in[5];   // [8,512]
    const float* estr = (const float*)d_in[6];   // [8]
    const float* eori = (const float*)d_in[7];   // [8]
    float* out = (float*)d_out;                  // [8,2048,2048]

    float* FQ = (float*)d_ws;                           // [8*2048, 12]
    float* FK = FQ + (size_t)BATCH * SEQ * KP;          // [8*2048, 12]

    // Stage 1: 2*8*2048 = 32768 tokens, 8 waves (tokens) per 256-thread block
    hts_feat_kernel<<<(2 * BATCH * SEQ) / 8, 256, 0, stream>>>(
        Q, Kx, ctr, orad, hrad, epts, estr, eori, FQ, FK);

    // Stage 2: 16384 strips of 16x128, 8 waves per block
    hts_gemm_kernel<<<16384 / 8, 256, 0, stream>>>(FQ, FK, out);
}